// VoxelNeXT_90915867722085
// MI455X (gfx1250) — compile-verified
//
#include <hip/hip_runtime.h>
#include <hip/hip_bf16.h>

// ---------------------------------------------------------------------------
// MI455X / gfx1250 VoxelNeXt-style sparse CNN. All channel-mixing math runs
// on V_WMMA_F32_16X16X4_F32 (fp32 matrix pipe). A wave32 owns a 16x16..16x64
// D tile. Gather masks are branch-free (clamped addresses + cndmask/mul);
// stores use a scalar-uniform full-tile fast path; per-lane addressing is
// 32-bit offsets off scalar bases (saddr + voffset form).
// ---------------------------------------------------------------------------

typedef __attribute__((ext_vector_type(2))) float v2f;
typedef __attribute__((ext_vector_type(8))) float v8f;

__device__ __forceinline__ v8f wmma_f32(v2f a, v2f b, v8f c) {
  return __builtin_amdgcn_wmma_f32_16x16x4_f32(false, a, false, b, (short)0, c,
                                               false, false);
}

__device__ __forceinline__ float gelu_exact(float x) {
  return 0.5f * x * (1.0f + erff(x * 0.70710678118654752f));
}

__device__ __forceinline__ int imin(int a, int b) { return a < b ? a : b; }
__device__ __forceinline__ int imax(int a, int b) { return a > b ? a : b; }

// ---------------------------------------------------------------------------
// Grid build: dense int32 lookup table, -1 = empty (memset 0xFF on host side).
// ---------------------------------------------------------------------------
__global__ void build_grid_k(const int* __restrict__ coords, int n,
                             int* __restrict__ grid, int Z, int Yd, int Xd) {
  int i = blockIdx.x * blockDim.x + threadIdx.x;
  if (i >= n) return;
  int b = coords[4 * i + 0], z = coords[4 * i + 1];
  int y = coords[4 * i + 2], x = coords[4 * i + 3];
  grid[((b * Z + z) * Yd + y) * Xd + x] = i;
}

// ---------------------------------------------------------------------------
// Input submanifold conv: 27 taps, Cin=5 (padded to 8), Cout=16.
// Weights staged zero-padded (27x8x16) in LDS: B fragments are bare ds loads.
// ---------------------------------------------------------------------------
__global__ void conv_in_wmma_k(const float* __restrict__ F5,
                               const int* __restrict__ coords,
                               const int* __restrict__ grid,
                               const float* __restrict__ W,  // [27][5][16]
                               float* __restrict__ Y, int N) {
  const int Z = 33, Yd = 256, Xd = 256;
  __shared__ float wlds[27 * 8 * 16];
  for (int i = threadIdx.x; i < 27 * 8 * 16; i += 128) {
    int t = i >> 7;         // tap (8*16 = 128 per tap)
    int r = i & 127;        // k*16+m
    int k = r >> 4, mm = r & 15;
    wlds[i] = (k < 5) ? W[(t * 5 + k) * 16 + mm] : 0.f;
  }
  __syncthreads();
  int lane = threadIdx.x & 31;
  int wv = __builtin_amdgcn_readfirstlane(threadIdx.x >> 5);
  int tileM = (blockIdx.x * 4 + wv) * 16;
  if (tileM >= N) return;  // scalar-uniform: EXEC stays all-ones for WMMA
  int m = lane & 15, hi = lane >> 4;
  int row = tileM + m;
  bool rv = row < N;
  int rowc = rv ? row : N - 1;
  int cb = coords[4 * rowc + 0], cz = coords[4 * rowc + 1];
  int cy = coords[4 * rowc + 2], cx = coords[4 * rowc + 3];
  int k0 = hi * 2;                 // {0,2}; +4 for second WMMA -> {4,6}
  int k0c = imin(k0 + 4, 4);       // clamp for A loads in the padded half
  v8f acc = {0.f, 0.f, 0.f, 0.f, 0.f, 0.f, 0.f, 0.f};
  int tap = 0;
  for (int dz = -1; dz <= 1; ++dz) {
    int nz = cz + dz;
    for (int dy = -1; dy <= 1; ++dy) {
      int ny = cy + dy;
      for (int dx = -1; dx <= 1; ++dx, ++tap) {
        int nx = cx + dx;
        bool ok = rv & (nz >= 0) & (nz < Z) & (ny >= 0) & (ny < Yd) &
                  (nx >= 0) & (nx < Xd);
        unsigned gaddr =
            (unsigned)(((cb * Z + imax(imin(nz, Z - 1), 0)) * Yd +
                        imax(imin(ny, Yd - 1), 0)) * Xd +
                       imax(imin(nx, Xd - 1), 0));
        int idx = grid[gaddr];  // unconditional load
        idx = ok ? idx : -1;    // cndmask
        float mk = idx >= 0 ? 1.f : 0.f;
        unsigned aoff = (unsigned)imax(idx, 0) * 5u;
        const float* wt = wlds + tap * 128;
        // kb = 0: K = {0,1,2,3} all valid in A
        v2f a0, b0;
        a0[0] = F5[aoff + k0] * mk;
        a0[1] = F5[aoff + k0 + 1] * mk;
        b0[0] = wt[k0 * 16 + m];
        b0[1] = wt[(k0 + 1) * 16 + m];
        acc = wmma_f32(a0, b0, acc);
        // kb = 4: only K=4 valid in A (hi==0 lane half); B padded with zeros
        float m0 = (k0 + 4 < 5) ? mk : 0.f;
        v2f a1, b1;
        a1[0] = F5[aoff + k0c] * m0;
        a1[1] = 0.f;
        b1[0] = wt[(k0 + 4) * 16 + m];
        b1[1] = wt[(k0 + 5) * 16 + m];
        acc = wmma_f32(a1, b1, acc);
      }
    }
  }
  if (tileM + 16 <= N) {  // scalar-uniform full-tile fast path
#pragma unroll
    for (int r = 0; r < 8; ++r)
      Y[(unsigned)(tileM + r + hi * 8) * 16u + m] = acc[r];
  } else {
#pragma unroll
    for (int r = 0; r < 8; ++r) {
      int ro = tileM + r + hi * 8;
      if (ro < N) Y[(unsigned)ro * 16u + m] = acc[r];
    }
  }
}

// ---------------------------------------------------------------------------
// Dense row-major GEMM: Y = X[NxK] @ W[KxCout], fused epilogues.
// EPI: 0 none, 1 +bias, 2 +bias+GELU, 3 residual + (acc+bias)*gamma.
// NT: N-tiles per wave (Cout must be divisible by 16*NT).
// ---------------------------------------------------------------------------
template <int EPI, int NT>
__global__ void gemm_wmma_k(const float* __restrict__ X,
                            const float* __restrict__ W,
                            const float* __restrict__ bias,
                            const float* __restrict__ gamma, float* Y,
                            const float* resid, int N, int K, int Cout) {
  int lane = threadIdx.x & 31;
  int wv = __builtin_amdgcn_readfirstlane(threadIdx.x >> 5);
  int tileM = (blockIdx.x * 4 + wv) * 16;
  if (tileM >= N) return;
  int tileN = blockIdx.y * (16 * NT);
  int m = lane & 15, hi = lane >> 4;
  int row = tileM + m;
  int rowc = row < N ? row : N - 1;
  int col = tileN + m;
  unsigned aoff = (unsigned)rowc * (unsigned)K + (unsigned)(hi * 2);
  unsigned woff = (unsigned)(hi * 2) * (unsigned)Cout + (unsigned)col;
  v8f acc[NT];
#pragma unroll
  for (int t = 0; t < NT; ++t)
    acc[t] = (v8f){0.f, 0.f, 0.f, 0.f, 0.f, 0.f, 0.f, 0.f};
#pragma unroll 4
  for (int k = 0; k < K; k += 4) {
    v2f a = *(const v2f*)(X + aoff);  // global_load_b64, 8B aligned
    aoff += 4u;
#pragma unroll
    for (int t = 0; t < NT; ++t) {
      v2f b;
      b[0] = W[woff + t * 16];
      b[1] = W[woff + (unsigned)Cout + t * 16];
      acc[t] = wmma_f32(a, b, acc[t]);
    }
    woff += 4u * (unsigned)Cout;
  }
  bool full = (tileM + 16 <= N);  // scalar-uniform
#pragma unroll
  for (int t = 0; t < NT; ++t) {
    int ct = col + t * 16;
    float bv = (EPI >= 1) ? bias[ct] : 0.f;
    float gv = (EPI == 3) ? gamma[ct] : 0.f;
    if (full) {
#pragma unroll
      for (int r = 0; r < 8; ++r) {
        unsigned ro = (unsigned)(tileM + r + hi * 8);
        float v = acc[t][r];
        if (EPI >= 1) v += bv;
        if (EPI == 2) v = gelu_exact(v);
        if (EPI == 3) v = resid[ro * (unsigned)Cout + ct] + v * gv;
        Y[ro * (unsigned)Cout + ct] = v;
      }
    } else {
#pragma unroll
      for (int r = 0; r < 8; ++r) {
        int ro = tileM + r + hi * 8;
        if (ro < N) {
          float v = acc[t][r];
          if (EPI >= 1) v += bv;
          if (EPI == 2) v = gelu_exact(v);
          if (EPI == 3) v = resid[(unsigned)ro * Cout + ct] + v * gv;
          Y[(unsigned)ro * Cout + ct] = v;
        }
      }
    }
  }
}

// ---------------------------------------------------------------------------
// Strided sparse downsample conv: gather-GEMM over kz*ky*kx taps, 16x32/wave.
// Cout divisible by 32; tap offsets non-negative.
// ---------------------------------------------------------------------------
__global__ void down_wmma_k(const float* __restrict__ X,
                            const int* __restrict__ oc,
                            const int* __restrict__ grid,
                            const float* __restrict__ W,  // [KT][Cin][Cout]
                            float* __restrict__ Y, int Nout, int Cin, int Cout,
                            int kz, int ky, int kx, int sz, int sy, int sx,
                            int Z, int Yd, int Xd) {
  int lane = threadIdx.x & 31;
  int wv = __builtin_amdgcn_readfirstlane(threadIdx.x >> 5);
  int tileM = (blockIdx.x * 4 + wv) * 16;
  if (tileM >= Nout) return;
  int tileN = blockIdx.y * 32;
  int m = lane & 15, hi = lane >> 4;
  int row = tileM + m;
  bool rv = row < Nout;
  int rowc = rv ? row : Nout - 1;
  int cb = oc[4 * rowc + 0];
  int cz = oc[4 * rowc + 1] * sz;
  int cy = oc[4 * rowc + 2] * sy;
  int cx = oc[4 * rowc + 3] * sx;
  int col = tileN + m;
  v8f acc0 = {0.f, 0.f, 0.f, 0.f, 0.f, 0.f, 0.f, 0.f};
  v8f acc1 = acc0;
  unsigned wtap = (unsigned)(hi * 2) * (unsigned)Cout + (unsigned)col;
  const unsigned wstep = (unsigned)Cin * (unsigned)Cout;
  for (int tz = 0; tz < kz; ++tz) {
    int nz = cz + tz;
    for (int ty = 0; ty < ky; ++ty) {
      int ny = cy + ty;
      for (int tx = 0; tx < kx; ++tx, wtap += wstep) {
        int nx = cx + tx;
        bool ok = rv & (nz < Z) & (ny < Yd) & (nx < Xd);
        unsigned gaddr = (unsigned)(((cb * Z + imin(nz, Z - 1)) * Yd +
                                     imin(ny, Yd - 1)) * Xd + imin(nx, Xd - 1));
        int idx = grid[gaddr];
        idx = ok ? idx : -1;
        float mk = idx >= 0 ? 1.f : 0.f;
        unsigned aoff =
            (unsigned)imax(idx, 0) * (unsigned)Cin + (unsigned)(hi * 2);
        unsigned woff = wtap;
#pragma unroll 4
        for (int k = 0; k < Cin; k += 4) {
          v2f a = *(const v2f*)(X + aoff);
          aoff += 4u;
          a[0] *= mk;
          a[1] *= mk;
          v2f b0, b1;
          b0[0] = W[woff];
          b0[1] = W[woff + (unsigned)Cout];
          b1[0] = W[woff + 16u];
          b1[1] = W[woff + (unsigned)Cout + 16u];
          woff += 4u * (unsigned)Cout;
          acc0 = wmma_f32(a, b0, acc0);
          acc1 = wmma_f32(a, b1, acc1);
        }
      }
    }
  }
  if (tileM + 16 <= Nout) {  // scalar-uniform full-tile fast path
#pragma unroll
    for (int r = 0; r < 8; ++r) {
      unsigned ro = (unsigned)(tileM + r + hi * 8);
      Y[ro * (unsigned)Cout + col] = acc0[r];
      Y[ro * (unsigned)Cout + col + 16] = acc1[r];
    }
  } else {
#pragma unroll
    for (int r = 0; r < 8; ++r) {
      int ro = tileM + r + hi * 8;
      if (ro < Nout) {
        Y[(unsigned)ro * Cout + col] = acc0[r];
        Y[(unsigned)ro * Cout + col + 16] = acc1[r];
      }
    }
  }
}

// ---------------------------------------------------------------------------
// Fused depthwise 3x9x9 conv + bias + LayerNorm, templated on C.
// One wave per point; uniform work (coords, bounds, grid lookup) scalarized
// via readfirstlane; lanes split channels; wave32 butterfly reductions.
// ---------------------------------------------------------------------------
template <int C>
__global__ void dw_ln_k(const float* __restrict__ X,
                        const int* __restrict__ coords,
                        const int* __restrict__ grid,
                        const float* __restrict__ dww,  // [243][C]
                        const float* __restrict__ dwb,
                        const float* __restrict__ lng,
                        const float* __restrict__ lnb, float* __restrict__ Y,
                        int N, int Z, int Yd, int Xd) {
  constexpr int NCH = (C + 31) / 32;
  int lane = threadIdx.x & 31;
  int wv = __builtin_amdgcn_readfirstlane(threadIdx.x >> 5);
  int p = blockIdx.x * 4 + wv;
  if (p >= N) return;
  int cb = __builtin_amdgcn_readfirstlane(coords[4 * p + 0]);
  int cz = __builtin_amdgcn_readfirstlane(coords[4 * p + 1]);
  int cy = __builtin_amdgcn_readfirstlane(coords[4 * p + 2]);
  int cx = __builtin_amdgcn_readfirstlane(coords[4 * p + 3]);
  int chc[NCH];
  float mk[NCH];
#pragma unroll
  for (int j = 0; j < NCH; ++j) {
    int ch = lane + j * 32;
    chc[j] = ch < C ? ch : C - 1;
    mk[j] = ch < C ? 1.f : 0.f;
  }
  float acc[NCH];
#pragma unroll
  for (int j = 0; j < NCH; ++j) acc[j] = 0.f;
  int tap = 0;
  for (int dz = -1; dz <= 1; ++dz) {
    int nz = cz + dz;
    bool okz = (nz >= 0) & (nz < Z);
    for (int dy = -4; dy <= 4; ++dy) {
      int ny = cy + dy;
      bool oky = okz & (ny >= 0) & (ny < Yd);
      int rowbase = ((cb * Z + imax(imin(nz, Z - 1), 0)) * Yd +
                     imax(imin(ny, Yd - 1), 0)) * Xd;
      for (int dx = -4; dx <= 4; ++dx, ++tap) {
        int nx = cx + dx;
        if (!(oky & (nx >= 0) & (nx < Xd))) continue;  // scalar-uniform skip
        int idx = __builtin_amdgcn_readfirstlane(grid[rowbase + nx]);
        if (idx < 0) continue;  // scalar-uniform skip
        unsigned xoff = (unsigned)idx * (unsigned)C;
        unsigned woff = (unsigned)tap * (unsigned)C;
#pragma unroll
        for (int j = 0; j < NCH; ++j)
          acc[j] += X[xoff + chc[j]] * dww[woff + chc[j]];
      }
    }
  }
  float vals[NCH];
  float s = 0.f;
#pragma unroll
  for (int j = 0; j < NCH; ++j) {
    float v = (acc[j] + dwb[chc[j]]) * mk[j];
    vals[j] = v;
    s += v;
  }
  for (int o = 16; o > 0; o >>= 1) s += __shfl_xor(s, o, 32);
  float mean = s / (float)C;
  float vs = 0.f;
#pragma unroll
  for (int j = 0; j < NCH; ++j) {
    float d = (vals[j] - mean) * mk[j];
    vs += d * d;
  }
  for (int o = 16; o > 0; o >>= 1) vs += __shfl_xor(vs, o, 32);
  float inv = rsqrtf(vs / (float)C + 1e-5f);
#pragma unroll
  for (int j = 0; j < NCH; ++j) {
    int ch = lane + j * 32;
    if (ch < C)
      Y[(unsigned)p * (unsigned)C + ch] =
          (vals[j] - mean) * inv * lng[ch] + lnb[ch];
  }
}

// ---------------------------------------------------------------------------
// BatchNorm (biased var over all rows) two-pass: stats + apply(ReLU).
// C is always a power of two (16..128).
// ---------------------------------------------------------------------------
__global__ void bn_stats_k(const float* __restrict__ X, int N, int C,
                           float* __restrict__ stats) {
  int c = threadIdx.x;  // blockDim.x == C
  float s = 0.f, q = 0.f;
  for (int r = blockIdx.x; r < N; r += gridDim.x) {
    float v = X[(unsigned)r * (unsigned)C + c];
    s += v;
    q += v * v;
  }
  atomicAdd(&stats[c], s);
  atomicAdd(&stats[C + c], q);
}

__global__ void bn_apply_k(float* __restrict__ X, int N, int C,
                           const float* __restrict__ stats,
                           const float* __restrict__ g,
                           const float* __restrict__ b) {
  unsigned i = blockIdx.x * blockDim.x + threadIdx.x;
  unsigned total = (unsigned)N * (unsigned)C;
  if (i >= total) return;
  int c = (int)(i & (unsigned)(C - 1));  // C is a power of two
  float mean = stats[c] / (float)N;
  float var = stats[C + c] / (float)N - mean * mean;
  float y = (X[i] - mean) * rsqrtf(var + 1e-3f) * g[c] + b[c];
  X[i] = y > 0.f ? y : 0.f;
}

// ---------------------------------------------------------------------------
// Host driver
// ---------------------------------------------------------------------------
extern "C" void kernel_launch(void* const* d_in, const int* in_sizes, int n_in,
                              void* d_out, int out_size, void* d_ws,
                              size_t ws_size, hipStream_t stream) {
  auto fp = [&](int i) { return (const float*)d_in[i]; };
  auto ip = [&](int i) { return (const int*)d_in[i]; };

  // -------- inputs (setup_inputs() dict insertion order, params recursed) --
  const float* vf = fp(0);
  int q = 1;
  const float* w_in = fp(q++);
  const float* bn_in_g = fp(q++);
  const float* bn_in_b = fp(q++);
  struct XT {
    const float *dw_w, *dw_b, *ln_g, *ln_b, *pw1_w, *pw1_b, *pw2_w, *pw2_b,
        *gamma;
  };
  auto ld_xt = [&]() {
    XT t;
    t.dw_w = fp(q++); t.dw_b = fp(q++); t.ln_g = fp(q++); t.ln_b = fp(q++);
    t.pw1_w = fp(q++); t.pw1_b = fp(q++); t.pw2_w = fp(q++); t.pw2_b = fp(q++);
    t.gamma = fp(q++);
    return t;
  };
  XT xt1 = ld_xt();
  const float* w_d2 = fp(q++); const float* bn2_g = fp(q++); const float* bn2_b = fp(q++);
  XT xt2 = ld_xt();
  const float* w_d3 = fp(q++); const float* bn3_g = fp(q++); const float* bn3_b = fp(q++);
  XT xt3 = ld_xt();
  const float* w_d4 = fp(q++); const float* bn4_g = fp(q++); const float* bn4_b = fp(q++);
  XT xt4 = ld_xt();
  const float* w_out = fp(q++);
  const float* bn_out_g = fp(q++);
  const float* bn_out_b = fp(q++);
  const int* c0 = ip(q); int N0 = in_sizes[q] / 4; q++;
  const int* c1 = ip(q); int N1 = in_sizes[q] / 4; q++;
  const int* c2 = ip(q); int N2 = in_sizes[q] / 4; q++;
  const int* c3 = ip(q); int N3 = in_sizes[q] / 4; q++;
  const int* cO = ip(q); int Nout = in_sizes[q] / 4; q++;
  (void)n_in; (void)out_size; (void)ws_size;

  // -------- workspace bump allocator ---------------------------------------
  char* ws = (char*)d_ws;
  size_t off = 0;
  auto alloc = [&](size_t bytes) {
    void* p = ws + off;
    off = (off + bytes + 255) & ~(size_t)255;
    return p;
  };
  int* g0 = (int*)alloc((size_t)2 * 33 * 256 * 256 * 4);
  int* g1 = (int*)alloc((size_t)2 * 16 * 128 * 128 * 4);
  int* g2 = (int*)alloc((size_t)2 * 8 * 64 * 64 * 4);
  int* g3 = (int*)alloc((size_t)2 * 4 * 32 * 32 * 4);
  float* stats = (float*)alloc(1024 * sizeof(float));
  float* x0a = (float*)alloc((size_t)N0 * 16 * 4);
  float* x0b = (float*)alloc((size_t)N0 * 16 * 4);
  float* h0 = (float*)alloc((size_t)N0 * 64 * 4);
  float* x1a = (float*)alloc((size_t)N1 * 32 * 4);
  float* x1b = (float*)alloc((size_t)N1 * 32 * 4);
  float* h1 = (float*)alloc((size_t)N1 * 128 * 4);
  float* x2a = (float*)alloc((size_t)N2 * 64 * 4);
  float* x2b = (float*)alloc((size_t)N2 * 64 * 4);
  float* h2 = (float*)alloc((size_t)N2 * 256 * 4);
  float* x3a = (float*)alloc((size_t)N3 * 128 * 4);
  float* x3b = (float*)alloc((size_t)N3 * 128 * 4);
  float* h3 = (float*)alloc((size_t)N3 * 512 * 4);
  float* xout = (float*)d_out;  // [Nout][128]

  auto mblocks = [&](int n) { return ((n + 15) / 16 + 3) / 4; };

  auto build = [&](int* g, const int* c, int n, int Z, int Yd, int Xd) {
    hipMemsetAsync(g, 0xFF, (size_t)2 * Z * Yd * Xd * 4, stream);
    build_grid_k<<<dim3((n + 255) / 256), 256, 0, stream>>>(c, n, g, Z, Yd, Xd);
  };
  auto run_bn = [&](float* x, int N, int C, const float* g, const float* b) {
    hipMemsetAsync(stats, 0, (size_t)2 * C * sizeof(float), stream);
    bn_stats_k<<<dim3(256), dim3(C), 0, stream>>>(x, N, C, stats);
    long tot = (long)N * C;
    bn_apply_k<<<dim3((unsigned)((tot + 255) / 256)), 256, 0, stream>>>(
        x, N, C, stats, g, b);
  };
  auto run_xt = [&](float* x, float* tmp, float* hid, const int* coords,
                    const int* grid, int N, int C, int Z, int Yd, int Xd,
                    const XT& P) {
    dim3 dwg((N + 3) / 4);
    if (C == 16)
      dw_ln_k<16><<<dwg, 128, 0, stream>>>(x, coords, grid, P.dw_w, P.dw_b,
                                           P.ln_g, P.ln_b, tmp, N, Z, Yd, Xd);
    else if (C == 32)
      dw_ln_k<32><<<dwg, 128, 0, stream>>>(x, coords, grid, P.dw_w, P.dw_b,
                                           P.ln_g, P.ln_b, tmp, N, Z, Yd, Xd);
    else if (C == 64)
      dw_ln_k<64><<<dwg, 128, 0, stream>>>(x, coords, grid, P.dw_w, P.dw_b,
                                           P.ln_g, P.ln_b, tmp, N, Z, Yd, Xd);
    else
      dw_ln_k<128><<<dwg, 128, 0, stream>>>(x, coords, grid, P.dw_w, P.dw_b,
                                            P.ln_g, P.ln_b, tmp, N, Z, Yd, Xd);
    // pw1: hid = gelu(tmp @ pw1_w + b); Cout = 4C in {64..512} -> NT=4
    gemm_wmma_k<2, 4><<<dim3(mblocks(N), (4 * C) / 64), 128, 0, stream>>>(
        tmp, P.pw1_w, P.pw1_b, nullptr, hid, nullptr, N, C, 4 * C);
    // pw2: x = x + (hid @ pw2_w + b)*gamma; Cout = C
    if (C == 16)
      gemm_wmma_k<3, 1><<<dim3(mblocks(N), 1), 128, 0, stream>>>(
          hid, P.pw2_w, P.pw2_b, P.gamma, x, x, N, 4 * C, C);
    else if (C == 32)
      gemm_wmma_k<3, 2><<<dim3(mblocks(N), 1), 128, 0, stream>>>(
          hid, P.pw2_w, P.pw2_b, P.gamma, x, x, N, 4 * C, C);
    else
      gemm_wmma_k<3, 4><<<dim3(mblocks(N), C / 64), 128, 0, stream>>>(
          hid, P.pw2_w, P.pw2_b, P.gamma, x, x, N, 4 * C, C);
  };
  auto run_down = [&](const float* X, const int* oc, const int* grid,
                      const float* W, float* Y, int No, int Cin, int Cout,
                      int kz, int ky, int kx, int sz, int sy, int sx, int Z,
                      int Yd, int Xd) {
    down_wmma_k<<<dim3(mblocks(No), Cout / 32), 128, 0, stream>>>(
        X, oc, grid, W, Y, No, Cin, Cout, kz, ky, kx, sz, sy, sx, Z, Yd, Xd);
  };

  // -------- pipeline -------------------------------------------------------
  build(g0, c0, N0, 33, 256, 256);
  build(g1, c1, N1, 16, 128, 128);
  build(g2, c2, N2, 8, 64, 64);
  build(g3, c3, N3, 4, 32, 32);

  conv_in_wmma_k<<<dim3(mblocks(N0)), 128, 0, stream>>>(vf, c0, g0, w_in, x0a, N0);
  run_bn(x0a, N0, 16, bn_in_g, bn_in_b);
  run_xt(x0a, x0b, h0, c0, g0, N0, 16, 33, 256, 256, xt1);

  run_down(x0a, c1, g0, w_d2, x1a, N1, 16, 32, 2, 2, 2, 2, 2, 2, 33, 256, 256);
  run_bn(x1a, N1, 32, bn2_g, bn2_b);
  run_xt(x1a, x1b, h1, c1, g1, N1, 32, 16, 128, 128, xt2);

  run_down(x1a, c2, g1, w_d3, x2a, N2, 32, 64, 2, 2, 2, 2, 2, 2, 16, 128, 128);
  run_bn(x2a, N2, 64, bn3_g, bn3_b);
  run_xt(x2a, x2b, h2, c2, g2, N2, 64, 8, 64, 64, xt3);

  run_down(x2a, c3, g2, w_d4, x3a, N3, 64, 128, 2, 2, 2, 2, 2, 2, 8, 64, 64);
  run_bn(x3a, N3, 128, bn4_g, bn4_b);
  run_xt(x3a, x3b, h3, c3, g3, N3, 128, 4, 32, 32, xt4);

  run_down(x3a, cO, g3, w_out, xout, Nout, 128, 128, 2, 1, 1, 2, 1, 1, 4, 32, 32);
  run_bn(xout, Nout, 128, bn_out_g, bn_out_b);
}